// GINBlock_18184891531553
// MI455X (gfx1250) — compile-verified
//
#include <hip/hip_runtime.h>

typedef __attribute__((ext_vector_type(2))) float v2f;
typedef __attribute__((ext_vector_type(8))) float v8f;

#define DOUT 128
#define DIN 64
#define BN_EPS 1e-5f

// ---------------------------------------------------------------------------
// K0: agg = x  (GIN: h = x + sum_neighbors, eps=0), zero the stats buffers
// ---------------------------------------------------------------------------
__global__ void __launch_bounds__(256) init_kernel(const float* __restrict__ x,
                                                   float* __restrict__ agg,
                                                   float* __restrict__ stats,
                                                   int n64) {
    int i = blockIdx.x * blockDim.x + threadIdx.x;
    if (i < 512) stats[i] = 0.0f;
    if (i < n64) agg[i] = x[i];
}

// ---------------------------------------------------------------------------
// K1: scatter-add  agg[dst[e]] += x[src[e]]  (one wave per edge, 2 f32/lane)
// ---------------------------------------------------------------------------
__global__ void __launch_bounds__(256) scatter_add_kernel(const float* __restrict__ x,
                                                          const int* __restrict__ src,
                                                          const int* __restrict__ dst,
                                                          float* __restrict__ agg,
                                                          int E) {
    int t = blockIdx.x * blockDim.x + threadIdx.x;
    int e = t >> 5;            // wave-uniform -> scalar loads of indices
    int lane = t & 31;
    if (e >= E) return;
    int s = src[e];
    int d = dst[e];
    const float* xs = x + (size_t)s * DIN;
    float* ad = agg + (size_t)d * DIN;
    atomicAdd(&ad[lane], xs[lane]);
    atomicAdd(&ad[lane + 32], xs[lane + 32]);
}

// ---------------------------------------------------------------------------
// K2/K4: WMMA f32 GEMM  D = ReLU(affine(A) @ W + bias), fused column stats.
// One wave = 16 rows x 128 cols (8 tiles of 16x16), K stepped by 4 per WMMA.
// A-frag layout (ISA 7.12.2, 32-bit A 16x4): lanes 0-15 hold K=k,k+1 of row
// M=lane; lanes 16-31 hold K=k+2,k+3 of row M=lane-16  -> one v2f load.
// B-frag: lanes 0-15 hold B[k][col],B[k+1][col]; lanes 16-31 B[k+2..3][col].
// D tile: VGPR v -> rows v / v+8, col = lane&15 within tile.
// ---------------------------------------------------------------------------
template <int K, bool AFF>
__global__ void __launch_bounds__(256)
gemm_bias_relu_stats(const float* __restrict__ A, const float* __restrict__ W,
                     const float* __restrict__ bias,
                     const float* __restrict__ scaleA, const float* __restrict__ shiftA,
                     float* __restrict__ D,
                     float* __restrict__ colsum, float* __restrict__ colsumsq,
                     int N) {
    __shared__ float lds_sum[DOUT];
    __shared__ float lds_sq[DOUT];
    int tid = threadIdx.x;
    if (tid < DOUT) { lds_sum[tid] = 0.0f; lds_sq[tid] = 0.0f; }
    __syncthreads();

    int wave = tid >> 5;
    int lane = tid & 31;
    int lane16 = lane & 15;
    int hi = lane >> 4;                 // half-wave select (K pair / row+8)
    int rt = blockIdx.x * 8 + wave;     // 16-row tile index
    int row0 = rt * 16;

    int arow = row0 + lane16;
    if (arow >= N) arow = N - 1;        // clamp; OOB rows masked at store/stats
    const float* Arow = A + (size_t)arow * K;
    int koff = hi * 2;

    v8f acc[8];
    v8f zero = {0.f, 0.f, 0.f, 0.f, 0.f, 0.f, 0.f, 0.f};
#pragma unroll
    for (int c = 0; c < 8; ++c) acc[c] = zero;

    for (int k = 0; k < K; k += 4) {
        v2f a = *(const v2f*)(Arow + k + koff);
        if (AFF) {   // fold BatchNorm of previous layer into the A operand
            v2f s = *(const v2f*)(scaleA + k + koff);
            v2f t = *(const v2f*)(shiftA + k + koff);
            a = a * s + t;
        }
#pragma unroll
        for (int c = 0; c < 8; ++c) {
            int col = c * 16 + lane16;
            v2f b;
            b.x = W[(k + koff) * DOUT + col];
            b.y = W[(k + koff + 1) * DOUT + col];
            acc[c] = __builtin_amdgcn_wmma_f32_16x16x4_f32(
                /*neg_a=*/false, a, /*neg_b=*/false, b,
                /*c_mod=*/(short)0, acc[c], /*reuse_a=*/false, /*reuse_b=*/false);
        }
    }

    // bias + ReLU + store + per-column partial stats (masked for OOB rows)
#pragma unroll
    for (int c = 0; c < 8; ++c) {
        int col = c * 16 + lane16;
        float bc = bias[col];
        float s = 0.0f, sq = 0.0f;
#pragma unroll
        for (int v = 0; v < 8; ++v) {
            float h = fmaxf(acc[c][v] + bc, 0.0f);
            int row = row0 + hi * 8 + v;
            if (row < N) {
                D[(size_t)row * DOUT + col] = h;
                s += h;
                sq += h * h;
            }
        }
        atomicAdd(&lds_sum[col], s);   // ds_add_f32
        atomicAdd(&lds_sq[col], sq);
    }
    __syncthreads();
    if (tid < DOUT)
        atomicAdd(&colsum[tid], lds_sum[tid]);
    else if (tid < 2 * DOUT)
        atomicAdd(&colsumsq[tid - DOUT], lds_sq[tid - DOUT]);
}

// ---------------------------------------------------------------------------
// K3/K5: fold BN(train) batch stats into per-column affine scale/shift
// ---------------------------------------------------------------------------
__global__ void bn_finalize(const float* __restrict__ colsum,
                            const float* __restrict__ colsumsq,
                            const float* __restrict__ gamma,
                            const float* __restrict__ beta,
                            float* __restrict__ scale, float* __restrict__ shift,
                            int N) {
    int c = threadIdx.x;
    if (c < DOUT) {
        float inv = 1.0f / (float)N;
        float mean = colsum[c] * inv;
        float var = fmaxf(colsumsq[c] * inv - mean * mean, 0.0f);
        float rs = rsqrtf(var + BN_EPS);
        float sc = gamma[c] * rs;
        scale[c] = sc;
        shift[c] = beta[c] - mean * sc;
    }
}

// ---------------------------------------------------------------------------
// K6: in-place BN2 apply on d_out (float4 vectorized)
// ---------------------------------------------------------------------------
__global__ void __launch_bounds__(256) bn_apply(float4* __restrict__ out,
                                                const float* __restrict__ scale,
                                                const float* __restrict__ shift,
                                                int n4) {
    int i = blockIdx.x * blockDim.x + threadIdx.x;
    if (i >= n4) return;
    float4 v = out[i];
    int c = (i * 4) & (DOUT - 1);
    v.x = v.x * scale[c + 0] + shift[c + 0];
    v.y = v.y * scale[c + 1] + shift[c + 1];
    v.z = v.z * scale[c + 2] + shift[c + 2];
    v.w = v.w * scale[c + 3] + shift[c + 3];
    out[i] = v;
}

// ---------------------------------------------------------------------------
extern "C" void kernel_launch(void* const* d_in, const int* in_sizes, int n_in,
                              void* d_out, int out_size, void* d_ws, size_t ws_size,
                              hipStream_t stream) {
    const float* x   = (const float*)d_in[0];
    const int*   ei  = (const int*)d_in[1];   // [2,E]: src then dst
    const float* W1  = (const float*)d_in[2];
    const float* b1  = (const float*)d_in[3];
    const float* g1  = (const float*)d_in[4];
    const float* be1 = (const float*)d_in[5];
    const float* W2  = (const float*)d_in[6];
    const float* b2  = (const float*)d_in[7];
    const float* g2  = (const float*)d_in[8];
    const float* be2 = (const float*)d_in[9];

    int N = in_sizes[0] / DIN;     // 100000
    int E = in_sizes[1] / 2;       // 1250000
    const int* src = ei;
    const int* dst = ei + E;

    // workspace layout
    float* agg   = (float*)d_ws;                 // N*64
    float* h1    = agg + (size_t)N * DIN;        // N*128
    float* stats = h1 + (size_t)N * DOUT;        // 4*128 (sum1,sq1,sum2,sq2)
    float* aff   = stats + 4 * DOUT;             // 4*128 (scale1,shift1,scale2,shift2)
    float* out   = (float*)d_out;

    int n64 = N * DIN;
    init_kernel<<<(n64 + 255) / 256, 256, 0, stream>>>(x, agg, stats, n64);

    long long sthreads = (long long)E * 32;
    scatter_add_kernel<<<(unsigned)((sthreads + 255) / 256), 256, 0, stream>>>(
        x, src, dst, agg, E);

    int tiles = (N + 15) / 16;
    int gblocks = (tiles + 7) / 8;

    gemm_bias_relu_stats<DIN, false><<<gblocks, 256, 0, stream>>>(
        agg, W1, b1, nullptr, nullptr, h1, stats, stats + DOUT, N);

    bn_finalize<<<1, DOUT, 0, stream>>>(stats, stats + DOUT, g1, be1,
                                        aff, aff + DOUT, N);

    gemm_bias_relu_stats<DOUT, true><<<gblocks, 256, 0, stream>>>(
        h1, W2, b2, aff, aff + DOUT, out, stats + 2 * DOUT, stats + 3 * DOUT, N);

    bn_finalize<<<1, DOUT, 0, stream>>>(stats + 2 * DOUT, stats + 3 * DOUT, g2, be2,
                                        aff + 2 * DOUT, aff + 3 * DOUT, N);

    int n4 = N * DOUT / 4;
    bn_apply<<<(n4 + 255) / 256, 256, 0, stream>>>((float4*)out,
                                                   aff + 2 * DOUT, aff + 3 * DOUT, n4);
}